// Encoder_386547056896
// MI455X (gfx1250) — compile-verified
//
#include <hip/hip_runtime.h>

typedef __attribute__((ext_vector_type(2))) float v2f;
typedef __attribute__((ext_vector_type(8))) float v8f;

#define FEAT        256
#define TWOF        512
#define B_TOTAL     50000
#define NSAMP       10
#define LDS_STRIDE  516   // 512 + 4 pad dwords -> conflict-free b64 reads, 16B-aligned f4 writes

__global__ __launch_bounds__(512)
void sage_encoder_kernel(const float* __restrict__ features,
                         const float* __restrict__ weight,
                         const int*   __restrict__ nodes,
                         const int*   __restrict__ neigh,
                         float*       __restrict__ out)
{
    __shared__ float comb[16 * LDS_STRIDE];

    const int tid  = threadIdx.x;
    const int wave = tid >> 5;     // 0..15
    const int lane = tid & 31;
    const int n0   = blockIdx.x * 16;

    // ---------------- Phase 1: gather + mean into LDS ----------------
    // Wave w builds combined row w: [ self(256) | neigh_mean(256) ].
    {
        const int row  = wave;
        const int node = nodes[n0 + row];
        const float* selfp = features + (size_t)node * FEAT;
        const int* nrow = neigh + (size_t)(n0 + row) * NSAMP;
        int nidx[NSAMP];
        #pragma unroll
        for (int s = 0; s < NSAMP; ++s) nidx[s] = nrow[s];

        float* dst = comb + row * LDS_STRIDE;
        #pragma unroll
        for (int it = 0; it < 2; ++it) {
            const int col = 4 * lane + 128 * it;   // coalesced float4 per wave
            float4 sf = *(const float4*)(selfp + col);
            float ax = 0.f, ay = 0.f, az = 0.f, aw = 0.f;
            #pragma unroll
            for (int s = 0; s < NSAMP; ++s) {
                float4 nf = *(const float4*)(features + (size_t)nidx[s] * FEAT + col);
                ax += nf.x; ay += nf.y; az += nf.z; aw += nf.w;
            }
            *(float4*)(dst + col) = sf;
            float4 m; m.x = ax * 0.1f; m.y = ay * 0.1f; m.z = az * 0.1f; m.w = aw * 0.1f;
            *(float4*)(dst + FEAT + col) = m;
        }
    }
    __syncthreads();

    // ---------------- Phase 2: 16x16 tile per wave via f32 WMMA ----------------
    // out[m, n] = relu( sum_k W[m,k] * comb[n,k] ),  m in [16w,16w+16), n in [n0,n0+16)
    const int m0   = wave * 16;
    const int half = lane >> 4;   // 0 -> K+0/K+1 ; 1 -> K+2/K+3 (A/B VGPR layout)
    const int r    = lane & 15;   // A: M index, B: N index, D: N index

    const float* wrow = weight + (size_t)(m0 + r) * TWOF + 2 * half;  // A fragment source
    const float* brow = comb   + (size_t)r * LDS_STRIDE + 2 * half;   // B fragment source (LDS)

    v8f c = {};
    #pragma unroll 8
    for (int k = 0; k < TWOF; k += 4) {
        v2f a = *(const v2f*)(wrow + k);   // W[m0+r][k+2*half .. +1]
        v2f b = *(const v2f*)(brow + k);   // comb[r][k+2*half .. +1]
        c = __builtin_amdgcn_wmma_f32_16x16x4_f32(
                /*neg_a=*/false, a, /*neg_b=*/false, b,
                /*c_mod=*/(short)0, c, /*reuse_a=*/false, /*reuse_b=*/false);
    }

    // D layout: VGPR j holds M = j + 8*half, N = r. Fused ReLU, coalesced store.
    float* obase = out + (size_t)(m0 + 8 * half) * B_TOTAL + n0 + r;
    #pragma unroll
    for (int j = 0; j < 8; ++j) {
        float v = c[j];
        obase[(size_t)j * B_TOTAL] = v > 0.f ? v : 0.f;
    }
}

extern "C" void kernel_launch(void* const* d_in, const int* in_sizes, int n_in,
                              void* d_out, int out_size, void* d_ws, size_t ws_size,
                              hipStream_t stream) {
    const float* features = (const float*)d_in[0];   // [100000, 256] f32
    const float* weight   = (const float*)d_in[1];   // [256, 512]   f32
    const int*   nodes    = (const int*)d_in[2];     // [50000]      i32
    const int*   neigh    = (const int*)d_in[3];     // [50000, 10]  i32
    float*       out      = (float*)d_out;           // [256, 50000] f32

    dim3 grid(B_TOTAL / 16);   // 3125 workgroups
    dim3 block(512);           // 16 wave32 waves
    hipLaunchKernelGGL(sage_encoder_kernel, grid, block, 0, stream,
                       features, weight, nodes, neigh, out);
}